// BlurredMem_61950608278069
// MI455X (gfx1250) — compile-verified
//
#include <hip/hip_runtime.h>
#include <math.h>

// Problem constants (from reference): bs=128, W=256, N=2048, R=4
#define BS 128
#define W 256
#define N 2048
#define R 4
#define TWOW 512
#define EPSV 1e-8f

// Workspace layout (in floats)
#define WS_WM     0            // (BS*W)   tanh(m_t)
#define WS_ERASE  32768        // (BS*W)   softmax(e_t)
#define WS_GATED  65536        // (BS*2W)  [g*write_m, (1-g)*m_erased]
#define WS_WK     131072       // (BS*W)   write_key (relu), then normalized in place
#define WS_KN     163840       // (BS*R*W) normalized read keys

typedef __attribute__((ext_vector_type(2))) float v2f;
typedef __attribute__((ext_vector_type(4))) float v4f;
typedef __attribute__((ext_vector_type(8))) float v8f;
typedef __attribute__((ext_vector_type(4))) unsigned int u32x4;
typedef __attribute__((ext_vector_type(8))) int i32x8;
typedef __attribute__((ext_vector_type(4))) int i32x4;

// ---------------- block reduction helpers ----------------
__device__ __forceinline__ float block_reduce_max(float v, float* red, int tid, int nthr) {
    red[tid] = v; __syncthreads();
    for (int s = nthr >> 1; s > 0; s >>= 1) {
        if (tid < s) red[tid] = fmaxf(red[tid], red[tid + s]);
        __syncthreads();
    }
    float r = red[0]; __syncthreads();
    return r;
}
__device__ __forceinline__ float block_reduce_sum(float v, float* red, int tid, int nthr) {
    red[tid] = v; __syncthreads();
    for (int s = nthr >> 1; s > 0; s >>= 1) {
        if (tid < s) red[tid] += red[tid + s];
        __syncthreads();
    }
    float r = red[0]; __syncthreads();
    return r;
}

// ---------------- K1: elementwise prep per batch ----------------
__global__ __launch_bounds__(256) void k1_prep(
    const float* __restrict__ k_r, const float* __restrict__ m_t,
    const float* __restrict__ e_t, const float* __restrict__ m_er,
    const float* __restrict__ gW, const float* __restrict__ gb,
    float* __restrict__ ws) {
    __shared__ float red[256];
    const int b = blockIdx.x, w = threadIdx.x;

    float wm = tanhf(m_t[b * W + w]);
    float me = m_er[b * W + w];
    ws[WS_WM + b * W + w] = wm;

    // erase_vec = softmax(e_t) over W
    float e = e_t[b * W + w];
    float mx = block_reduce_max(e, red, w, 256);
    float ex = expf(e - mx);
    float es = block_reduce_sum(ex, red, w, 256);
    ws[WS_ERASE + b * W + w] = ex / es;

    // mix gate: g = sigmoid(cat . gW + gb), cat = [wm, me]
    float part = wm * gW[w] + me * gW[W + w];
    float tot = block_reduce_sum(part, red, w, 256);
    float g = 1.f / (1.f + expf(-(tot + gb[0])));
    ws[WS_GATED + b * TWOW + w]     = g * wm;
    ws[WS_GATED + b * TWOW + W + w] = (1.f - g) * me;

    // normalized read keys
    for (int r = 0; r < R; ++r) {
        float kv = tanhf(k_r[b * (R * W) + r * W + w]);
        float ss = block_reduce_sum(kv * kv, red, w, 256);
        ws[WS_KN + (b * R + r) * W + w] = kv / (sqrtf(ss) + EPSV);
    }
}

// ---------------- K2: write_key GEMM via V_WMMA_F32_16X16X4_F32 ----------------
// C(128x256) = gated(128x512) x oW^T(512x256), relu(+ob).  One wave per 16x16 tile.
__global__ __launch_bounds__(32) void k2_wkey_gemm(
    const float* __restrict__ ws, const float* __restrict__ oW,
    const float* __restrict__ ob, float* __restrict__ wk) {
    const int lane = threadIdx.x;
    const int half = lane >> 4, l16 = lane & 15;
    const int bm = blockIdx.y, bn = blockIdx.x;
    const int m = bm * 16 + l16;
    const int n = bn * 16 + l16;
    const float* G = ws + WS_GATED;

    v8f c = {};
    for (int k0 = 0; k0 < TWOW; k0 += 4) {
        const int kb = k0 + half * 2;
        v2f a  = *(const v2f*)(G  + (size_t)m * TWOW + kb);
        v2f bb = *(const v2f*)(oW + (size_t)n * TWOW + kb);
        c = __builtin_amdgcn_wmma_f32_16x16x4_f32(
            false, a, false, bb, (short)0, c, false, false);
    }
    for (int j = 0; j < 8; ++j) {
        const int row = bm * 16 + half * 8 + j;
        const int col = bn * 16 + l16;
        float v = c[j] + ob[col];
        wk[row * W + col] = v > 0.f ? v : 0.f;
    }
}

// ---------------- K2b: normalize write_key rows in place ----------------
__global__ __launch_bounds__(256) void k2b_norm_wk(float* __restrict__ wk) {
    __shared__ float red[256];
    const int b = blockIdx.x, w = threadIdx.x;
    float v = wk[b * W + w];
    float ss = block_reduce_sum(v * v, red, w, 256);
    wk[b * W + w] = v / (sqrtf(ss) + EPSV);
}

// ---------------- KMEM: fused write-dist softmax + M_new + read dists ----------------
// One block per batch; 512 threads; thread t owns columns 4t..4t+3 (b128 traffic).
__global__ __launch_bounds__(512) void kmem(
    const float* __restrict__ mem, const float* __restrict__ ws,
    float* __restrict__ Mnew, float* __restrict__ rdist) {
    __shared__ float s_wkn[W], s_er[W], s_wm[W], s_kn[R * W], s_red[512];
    const int b = blockIdx.x, t = threadIdx.x;
    if (t < W) {
        s_wkn[t] = ws[WS_WK + b * W + t];
        s_er[t]  = ws[WS_ERASE + b * W + t];
        s_wm[t]  = ws[WS_WM + b * W + t];
    }
    s_kn[t]       = ws[WS_KN + b * (R * W) + t];
    s_kn[t + 512] = ws[WS_KN + b * (R * W) + 512 + t];
    __syncthreads();

    const float* M = mem + (size_t)b * W * N + 4 * t;

    // Phase 1: write-key dissimilarity for 4 columns
    float dot[4] = {0.f, 0.f, 0.f, 0.f}, ss[4] = {0.f, 0.f, 0.f, 0.f};
    for (int w = 0; w < W; ++w) {
        v4f v = *(const v4f*)(M + (size_t)w * N);
        float wkv = s_wkn[w];
        #pragma unroll
        for (int j = 0; j < 4; ++j) { dot[j] += wkv * v[j]; ss[j] += v[j] * v[j]; }
    }
    float dist[4];
    #pragma unroll
    for (int j = 0; j < 4; ++j) dist[j] = -(dot[j] / (sqrtf(ss[j]) + EPSV));

    // softmax over N = 2048
    float lm = fmaxf(fmaxf(dist[0], dist[1]), fmaxf(dist[2], dist[3]));
    float mx = block_reduce_max(lm, s_red, t, 512);
    float ev[4]; float ls = 0.f;
    #pragma unroll
    for (int j = 0; j < 4; ++j) { ev[j] = expf(dist[j] - mx); ls += ev[j]; }
    float sum = block_reduce_sum(ls, s_red, t, 512);
    const float sinv = 1.f / sum;

    // Phase 2: M_new, column norms, 4 read-key dots (all vectorized over 4 cols)
    float* Mo = Mnew + (size_t)b * W * N + 4 * t;
    float wt[4];
    #pragma unroll
    for (int j = 0; j < 4; ++j) wt[j] = ev[j] * sinv;
    float ssn[4] = {0.f, 0.f, 0.f, 0.f};
    float acc[R][4] = {};
    for (int w = 0; w < W; ++w) {
        v4f v = *(const v4f*)(M + (size_t)w * N);
        float er = s_er[w], wmv = s_wm[w];
        v4f nv;
        #pragma unroll
        for (int j = 0; j < 4; ++j) nv[j] = v[j] * (1.f - er * wt[j]) + wmv * wt[j];
        *(v4f*)(Mo + (size_t)w * N) = nv;
        float k0 = s_kn[w], k1 = s_kn[W + w], k2 = s_kn[2 * W + w], k3 = s_kn[3 * W + w];
        #pragma unroll
        for (int j = 0; j < 4; ++j) {
            ssn[j]    += nv[j] * nv[j];
            acc[0][j] += k0 * nv[j];
            acc[1][j] += k1 * nv[j];
            acc[2][j] += k2 * nv[j];
            acc[3][j] += k3 * nv[j];
        }
    }
    float di[4];
    #pragma unroll
    for (int j = 0; j < 4; ++j) di[j] = 1.f / (sqrtf(ssn[j]) + EPSV);
    #pragma unroll
    for (int r = 0; r < R; ++r) {
        v4f o;
        #pragma unroll
        for (int j = 0; j < 4; ++j) o[j] = acc[r][j] * di[j];
        *(v4f*)(rdist + ((size_t)b * R + r) * N + 4 * t) = o;
    }
}

// ---------------- K5: softmax of read dists (in place, in d_out) ----------------
__global__ __launch_bounds__(256) void k5_read_softmax(float* __restrict__ rwt) {
    __shared__ float red[256];
    const int br = blockIdx.x, t = threadIdx.x;
    float* p = rwt + (size_t)br * N;
    float d[8], lm = -3.402823466e38f;
    for (int j = 0; j < 8; ++j) { d[j] = p[t + j * 256]; lm = fmaxf(lm, d[j]); }
    float mx = block_reduce_max(lm, red, t, 256);
    float ls = 0.f;
    for (int j = 0; j < 8; ++j) { d[j] = expf(d[j] - mx); ls += d[j]; }
    float s = block_reduce_sum(ls, red, t, 256);
    const float inv = 1.f / s;
    for (int j = 0; j < 8; ++j) p[t + j * 256] = d[j] * inv;
}

// ---------------- K6: m_read = read_wt @ M_new^T ----------------
// read_wt block (32KB) DMA'd into LDS by the Tensor Data Mover; each wave reduces
// one M_new row with b128 loads for all 4 heads simultaneously.
__global__ __launch_bounds__(256) void k6_mread(
    const float* __restrict__ Mnew, const float* __restrict__ rwt,
    float* __restrict__ mread) {
    __shared__ float s_wt[R * N];   // 32 KB
    const int b = blockIdx.x, t = threadIdx.x;

    if (t < 32) {
        // TDM descriptor: 1-row tile of 8192 fp32 (= R*N), global -> LDS.
        unsigned long long ga = (unsigned long long)(size_t)(rwt + (size_t)b * R * N);
        unsigned int lds = (unsigned int)(size_t)&s_wt[0];  // flat LDS addr low 32 = LDS offset
        u32x4 g0;
        g0[0] = 1u;                                          // count=1, user mode
        g0[1] = lds;                                         // lds_addr (bytes)
        g0[2] = (unsigned int)(ga & 0xffffffffu);            // global_addr[31:0]
        g0[3] = (unsigned int)((ga >> 32) & 0x1ffffffu)      // global_addr[56:32]
              | (2u << 30);                                  // type = 2 ("image")
        i32x8 g1;
        g1[0] = (int)(2u << 16);        // workgroup_mask=0, data_size=2 (4B)
        g1[1] = (int)(8192u << 16);     // tensor_dim0[15:0]=8192 in bits 79:48
        g1[2] = (int)(1u << 16);        // tensor_dim0 hi=0 | tensor_dim1=1
        g1[3] = (int)(8192u << 16);     // tensor_dim1 hi=0 | tile_dim0=8192
        g1[4] = 1;                      // tile_dim1=1, tile_dim2=0
        g1[5] = 8192;                   // tensor_dim0_stride[31:0]
        g1[6] = 0;                      // stride hi, tensor_dim1_stride lo
        g1[7] = 0;
        i32x4 z4 = {0, 0, 0, 0};
        i32x8 z8 = {0, 0, 0, 0, 0, 0, 0, 0};
        __builtin_amdgcn_tensor_load_to_lds(g0, g1, z4, z4, z8, 0);
        __builtin_amdgcn_s_wait_tensorcnt(0);
    }
    __syncthreads();

    const int wid = t >> 5, lane = t & 31;
    const float* Mb = Mnew + (size_t)b * W * N;
    for (int w = wid; w < W; w += 8) {
        const float* row = Mb + (size_t)w * N;
        float a0 = 0.f, a1 = 0.f, a2 = 0.f, a3 = 0.f;
        for (int n0 = lane * 4; n0 < N; n0 += 128) {
            v4f v  = *(const v4f*)(row + n0);
            v4f w0 = *(const v4f*)&s_wt[n0];
            v4f w1 = *(const v4f*)&s_wt[N + n0];
            v4f w2 = *(const v4f*)&s_wt[2 * N + n0];
            v4f w3 = *(const v4f*)&s_wt[3 * N + n0];
            #pragma unroll
            for (int j = 0; j < 4; ++j) {
                a0 += w0[j] * v[j];
                a1 += w1[j] * v[j];
                a2 += w2[j] * v[j];
                a3 += w3[j] * v[j];
            }
        }
        for (int off = 16; off > 0; off >>= 1) {
            a0 += __shfl_xor(a0, off, 32);
            a1 += __shfl_xor(a1, off, 32);
            a2 += __shfl_xor(a2, off, 32);
            a3 += __shfl_xor(a3, off, 32);
        }
        if (lane == 0) {
            mread[((size_t)b * R + 0) * W + w] = a0;
            mread[((size_t)b * R + 1) * W + w] = a1;
            mread[((size_t)b * R + 2) * W + w] = a2;
            mread[((size_t)b * R + 3) * W + w] = a3;
        }
    }
}

extern "C" void kernel_launch(void* const* d_in, const int* in_sizes, int n_in,
                              void* d_out, int out_size, void* d_ws, size_t ws_size,
                              hipStream_t stream) {
    const float* k_r    = (const float*)d_in[0];
    const float* m_t    = (const float*)d_in[1];
    const float* e_t    = (const float*)d_in[2];
    const float* m_er   = (const float*)d_in[3];
    const float* memory = (const float*)d_in[4];
    const float* gW     = (const float*)d_in[5];
    const float* gb     = (const float*)d_in[6];
    const float* oW     = (const float*)d_in[7];
    const float* ob     = (const float*)d_in[8];

    float* out   = (float*)d_out;
    float* mread = out;                                   // BS*R*W
    float* Mnew  = out + (size_t)BS * R * W;              // BS*W*N
    float* rwt   = out + (size_t)BS * R * W + (size_t)BS * W * N;  // BS*R*N
    float* ws    = (float*)d_ws;

    k1_prep<<<BS, 256, 0, stream>>>(k_r, m_t, e_t, m_er, gW, gb, ws);
    k2_wkey_gemm<<<dim3(W / 16, BS / 16), 32, 0, stream>>>(ws, oW, ob, ws + WS_WK);
    k2b_norm_wk<<<BS, 256, 0, stream>>>(ws + WS_WK);
    kmem<<<BS, 512, 0, stream>>>(memory, ws, Mnew, rwt);
    k5_read_softmax<<<BS * R, 256, 0, stream>>>(rwt);
    k6_mread<<<BS, 256, 0, stream>>>(Mnew, rwt, mread);
}